// LapLoss_74268574482988
// MI455X (gfx1250) — compile-verified
//
#include <hip/hip_runtime.h>

typedef __attribute__((ext_vector_type(2))) float v2f;
typedef __attribute__((ext_vector_type(8))) float v8f;

__device__ __forceinline__ int refl(int v, int n) {
    if (v < 0) v = -v;
    if (v >= n) v = 2 * n - 2 - v;
    return v;
}

// Wave32 sum via V_WMMA_F32_16X16X4_F32: lane partial -> A(16x4), B = ones,
// D rows = row sums. Each lane sums its 8 D components (rows 0..7 for lanes
// 0..15, rows 8..15 for lanes 16..31); one ds_swizzle SWAPX16 completes it.
// Requires EXEC all-ones (callers guarantee full blocks, uniform flow).
__device__ __forceinline__ float wave_sum32(float p) {
    v2f a; a.x = p;    a.y = 0.0f;
    v2f b; b.x = 1.0f; b.y = 1.0f;
    v8f c = {};
    v8f d = __builtin_amdgcn_wmma_f32_16x16x4_f32(false, a, false, b,
                                                  (short)0, c, false, false);
    float s = d[0] + d[1] + d[2] + d[3] + d[4] + d[5] + d[6] + d[7];
    // SWAPX16: group-of-32 swizzle, xor_mask=0x10, and_mask=0x1f
    s += __int_as_float(__builtin_amdgcn_ds_swizzle(__float_as_int(s), 0x401f));
    return s;
}

__device__ __forceinline__ float block_sum256(float p, float* red) {
    float w = wave_sum32(p);
    const int lane = threadIdx.x & 31;
    const int wid  = threadIdx.x >> 5;
    if (lane == 0) red[wid] = w;
    __syncthreads();
    float t = 0.0f;
    if (threadIdx.x == 0) {
        #pragma unroll
        for (int i = 0; i < 8; ++i) t += red[i];
    }
    return t; // valid on thread 0
}

// down[yd][xd] = (K5 (x) K5) * cur_reflectpad at (2yd,2xd); K5=[1,4,6,4,1]/16.
// Separable in LDS. Level 0: cur = A - B on the fly.
__global__ __launch_bounds__(256) void gauss_down_kernel(
    const float* __restrict__ A, const float* __restrict__ B, int isDiff,
    float* __restrict__ down, int H, int Hd)
{
    __shared__ float tin[20][68];
    __shared__ float hb[20][32];
    const int plane = blockIdx.z;
    const int xd0 = blockIdx.x * 32;
    const int yd0 = blockIdx.y * 8;
    const int tw   = min(32, Hd - xd0);
    const int gx0 = 2 * xd0 - 2;
    const int gy0 = 2 * yd0 - 2;
    const size_t poff = (size_t)plane * H * H;
    const float* __restrict__ pA = A + poff;

    if (tw == 32) {             // common case: compile-time cols = 68
        for (int idx = threadIdx.x; idx < 20 * 68; idx += 256) {
            const int r  = idx / 68;
            const int cc = idx - r * 68;
            const int gy = refl(gy0 + r, H);
            const int gx = refl(gx0 + cc, H);
            float v = pA[(size_t)gy * H + gx];
            if (isDiff) v -= B[poff + (size_t)gy * H + gx];
            tin[r][cc] = v;
        }
    } else {                    // only level 4 (Hd = 16)
        const int cols = 2 * tw + 4;
        for (int idx = threadIdx.x; idx < 20 * cols; idx += 256) {
            const int r  = idx / cols;
            const int cc = idx - r * cols;
            const int gy = refl(gy0 + r, H);
            const int gx = refl(gx0 + cc, H);
            float v = pA[(size_t)gy * H + gx];
            if (isDiff) v -= B[poff + (size_t)gy * H + gx];
            tin[r][cc] = v;
        }
    }
    __syncthreads();

    if (tw == 32) {
        for (int idx = threadIdx.x; idx < 20 * 32; idx += 256) {
            const int r  = idx >> 5;
            const int xd = idx & 31;
            const float* row = &tin[r][2 * xd];
            hb[r][xd] = 0.0625f * (row[0] + row[4]) + 0.25f * (row[1] + row[3])
                      + 0.375f * row[2];
        }
    } else {
        for (int idx = threadIdx.x; idx < 20 * tw; idx += 256) {
            const int r  = idx / tw;
            const int xd = idx - r * tw;
            const float* row = &tin[r][2 * xd];
            hb[r][xd] = 0.0625f * (row[0] + row[4]) + 0.25f * (row[1] + row[3])
                      + 0.375f * row[2];
        }
    }
    __syncthreads();

    const int xd = threadIdx.x & 31;
    const int yd = threadIdx.x >> 5;
    if (xd < tw) {
        const int r = 2 * yd;
        float v = 0.0625f * (hb[r][xd] + hb[r + 4][xd])
                + 0.25f   * (hb[r + 1][xd] + hb[r + 3][xd])
                + 0.375f  *  hb[r + 2][xd];
        down[(size_t)plane * Hd * Hd + (size_t)(yd0 + yd) * Hd + (xd0 + xd)] = v;
    }
}

// up = blur(zero-stuff(4*down)): parity of the stuffed coordinate selects the
// tap pattern (even: {|h-1|, h, min(h+1,Hd-1)} w={1,6,1}/16; odd:
// {h, min(h+1,Hd-1), 0} w={4,4,0}/16). Fully branchless 3x3 gather.
// val = |cur - up|; block-reduced to one partial per block (no global atomics).
__global__ __launch_bounds__(256) void uplap_kernel(
    const float* __restrict__ A, const float* __restrict__ B, int isDiff,
    const float* __restrict__ down, float* __restrict__ partials, int H)
{
    __shared__ float red[8];
    const int plane = blockIdx.z;
    const int x = blockIdx.x * 32 + (threadIdx.x & 31);
    const int y = blockIdx.y * 8  + (threadIdx.x >> 5);
    const size_t poff = (size_t)plane * H * H;

    float cur = A[poff + (size_t)y * H + x];
    if (isDiff) cur -= B[poff + (size_t)y * H + x];

    const int Hd = H >> 1;
    const float* __restrict__ dw = down + (size_t)plane * Hd * Hd;

    const int  yh = y >> 1,          xh = x >> 1;
    const bool ye = ((y & 1) == 0),  xe = ((x & 1) == 0);
    const int  yc = min(yh + 1, Hd - 1), xc = min(xh + 1, Hd - 1);

    const int iy0 = ye ? (yh == 0 ? 1 : yh - 1) : yh;
    const int iy1 = ye ? yh : yc;
    const int iy2 = ye ? yc : 0;
    const float wy0 = ye ? 0.0625f : 0.25f;
    const float wy1 = ye ? 0.375f  : 0.25f;
    const float wy2 = ye ? 0.0625f : 0.0f;

    const int ix0 = xe ? (xh == 0 ? 1 : xh - 1) : xh;
    const int ix1 = xe ? xh : xc;
    const int ix2 = xe ? xc : 0;
    const float wx0 = xe ? 0.0625f : 0.25f;
    const float wx1 = xe ? 0.375f  : 0.25f;
    const float wx2 = xe ? 0.0625f : 0.0f;

    const float* r0 = dw + (size_t)iy0 * Hd;
    const float* r1 = dw + (size_t)iy1 * Hd;
    const float* r2 = dw + (size_t)iy2 * Hd;
    const float s0 = wx0 * r0[ix0] + wx1 * r0[ix1] + wx2 * r0[ix2];
    const float s1 = wx0 * r1[ix0] + wx1 * r1[ix1] + wx2 * r1[ix2];
    const float s2 = wx0 * r2[ix0] + wx1 * r2[ix1] + wx2 * r2[ix2];
    const float up = 4.0f * (wy0 * s0 + wy1 * s1 + wy2 * s2);

    const float val = fabsf(cur - up);
    const float tot = block_sum256(val, red);
    if (threadIdx.x == 0) {
        const int bid = blockIdx.x + gridDim.x * (blockIdx.y + gridDim.y * blockIdx.z);
        partials[bid] = tot;
    }
}

__global__ __launch_bounds__(256) void final_kernel(
    const float* __restrict__ pb, float* __restrict__ out)
{
    __shared__ float red[8];
    const int   counts[5] = {49152, 12288, 3072, 768, 192};
    const float coef[5] = {
        1.0f  / (48.0f * 512.0f * 512.0f),
        2.0f  / (48.0f * 256.0f * 256.0f),
        4.0f  / (48.0f * 128.0f * 128.0f),
        8.0f  / (48.0f *  64.0f *  64.0f),
        16.0f / (48.0f *  32.0f *  32.0f)};
    float acc = 0.0f;
    const float* p = pb;
    for (int l = 0; l < 5; ++l) {
        float s = 0.0f;
        for (int i = threadIdx.x; i < counts[l]; i += 256) s += p[i];
        acc += s * coef[l];
        p += counts[l];
    }
    const float tot = block_sum256(acc, red);
    if (threadIdx.x == 0) out[0] = tot;
}

extern "C" void kernel_launch(void* const* d_in, const int* in_sizes, int n_in,
                              void* d_out, int out_size, void* d_ws, size_t ws_size,
                              hipStream_t stream) {
    (void)in_sizes; (void)n_in; (void)out_size; (void)ws_size;
    const float* p = (const float*)d_in[0];
    const float* t = (const float*)d_in[1];
    float* ws  = (float*)d_ws;
    float* out = (float*)d_out;

    const int S[5] = {512, 256, 128, 64, 32};

    float* dbuf[5];
    size_t off = 0;
    for (int l = 0; l < 5; ++l) {
        dbuf[l] = ws + off;
        off += (size_t)48 * (S[l] / 2) * (S[l] / 2);
    }
    float* pbase = ws + off;
    float* parts[5];
    for (int l = 0; l < 5; ++l) {
        parts[l] = ws + off;
        off += (size_t)48 * S[l] * S[l] / 256;   // one partial per 256-thread block
    }

    for (int l = 0; l < 5; ++l) {
        const float* A = (l == 0) ? p : dbuf[l - 1];
        const float* B = (l == 0) ? t : nullptr;
        const int isDiff = (l == 0) ? 1 : 0;
        const int H = S[l], Hd = H / 2;

        dim3 gd((Hd + 31) / 32, Hd / 8, 48);
        gauss_down_kernel<<<gd, 256, 0, stream>>>(A, B, isDiff, dbuf[l], H, Hd);

        dim3 gu(H / 32, H / 8, 48);
        uplap_kernel<<<gu, 256, 0, stream>>>(A, B, isDiff, dbuf[l], parts[l], H);
    }
    final_kernel<<<1, 256, 0, stream>>>(pbase, out);
}